// MambaBlock_25048249270575
// MI455X (gfx1250) — compile-verified
//
#include <hip/hip_runtime.h>
#include <hip/hip_bf16.h>

typedef __attribute__((ext_vector_type(2))) float v2f;
typedef __attribute__((ext_vector_type(8))) float v8f;

#define BATCH   2
#define SEQ     2048
#define DMODEL  1024
#define DINNER  2048
#define DSTATE  16
#define DTRANK  64
#define DCONV   4
#define LNEPS   1e-5f
#define SCHUNK  128

// ---------------------------------------------------------------------------
// WMMA fp32 GEMM:  C[M,N] = A[M,K] @ B[K,N], all strides compile-time.
// One wave -> one 16x16 tile (V_WMMA_F32_16X16X4_F32); 8 waves/block = 128 cols.
// A tile (16 x KC) staged in LDS once per block chunk (8x reuse across waves),
// bank-conflict-free via +4 float row padding. B loads get immediate offsets.
// EPI: 0 = none, 1 = softplus(val + bias[col]).
// Fragment layouts (CDNA5 ISA 7.12.2, f32):
//   A (16x4):  lane = m + 16*(k>=2), vgpr = k&1
//   B (4x16):  lane = n + 16*(k>=2), vgpr = k&1   (N<->M mirror of A)
//   C/D:       vgpr v holds row (v + 8*(lane>=16)), col = lane&15
// ---------------------------------------------------------------------------
template <int K, int LDA, int LDB, int LDC, int EPI>
__global__ void wmma_gemm_f32(const float* __restrict__ A, const float* __restrict__ Bm,
                              float* __restrict__ C, const float* __restrict__ bias,
                              int N)
{
  constexpr int KC  = 64;        // K-chunk staged in LDS
  constexpr int SAS = KC + 4;    // padded LDS row stride (bank-conflict free)
  __shared__ __align__(16) float sA[16 * SAS];

  const int tid   = threadIdx.x;
  const int lane  = tid & 31;
  const int wave  = tid >> 5;
  const int half  = lane >> 4;   // 0: lanes 0-15, 1: lanes 16-31
  const int l15   = lane & 15;
  const int tileN = blockIdx.x * 128 + wave * 16;
  const int tileM = blockIdx.y * 16;
  const bool active = (tileN < N);            // wave-uniform guard

  // cooperative A-load mapping: 256 threads x float4 = 16 rows x 64 cols
  const int ar = tid >> 4;                    // A row 0..15
  const int ac = (tid & 15) * 4;              // A col group
  const float* Aload = A + (size_t)(tileM + ar) * LDA + ac;

  const float* Bcol = Bm + tileN + l15;       // this lane's N column
  const float* sArow = sA + l15 * SAS + half * 2;

  v8f acc = {0.f,0.f,0.f,0.f,0.f,0.f,0.f,0.f};

  for (int k0 = 0; k0 < K; k0 += KC) {
    // stage A[tileM..+16) x [k0..k0+KC) in LDS (coalesced b128)
    const float4 av = *(const float4*)(Aload + k0);
    *(float4*)(sA + ar * SAS + ac) = av;
    if (k0 + KC < K)
      __builtin_prefetch(Aload + k0 + KC, 0, 1);   // global_prefetch_b8
    __syncthreads();

    if (active) {
      const float* Bk = Bcol + (size_t)k0 * LDB;
      #pragma unroll
      for (int kk = 0; kk < KC; kk += 4) {
        const int ka = kk + half * 2;
        v2f a; a.x = sArow[kk];               // ds_load_b64, conflict-free
               a.y = sArow[kk + 1];
        v2f b; b.x = Bk[ka * LDB];            // immediate-offset global loads
               b.y = Bk[(ka + 1) * LDB];
        acc = __builtin_amdgcn_wmma_f32_16x16x4_f32(
            false, a, false, b, (short)0, acc, false, false);
      }
    }
    __syncthreads();
  }

  if (active) {
    union { v8f v; float f[8]; } r; r.v = acc;
    const int col = tileN + l15;
    #pragma unroll
    for (int v = 0; v < 8; ++v) {
      const int row = tileM + v + half * 8;
      float val = r.f[v];
      if (EPI == 1) {                          // softplus(x + b_dt)
        val += bias[col];
        val = (val > 20.f) ? val : log1pf(__expf(val));
      }
      C[(size_t)row * LDC + col] = val;
    }
  }
}

// ---------------------------------------------------------------------------
// Depthwise causal conv (d_conv=4, left pad 3) + bias + SiLU.
// xin = xz[..., 0:DINNER] with row stride 2*DINNER.
// ---------------------------------------------------------------------------
__global__ void conv_silu(const float* __restrict__ xz, const float* __restrict__ cw,
                          const float* __restrict__ cb, float* __restrict__ u)
{
  const size_t idx = (size_t)blockIdx.x * blockDim.x + threadIdx.x;
  const size_t total = (size_t)BATCH * SEQ * DINNER;
  if (idx >= total) return;
  const int d = (int)(idx % DINNER);
  const int t = (int)((idx / DINNER) % SEQ);
  const int b = (int)(idx / ((size_t)DINNER * SEQ));
  float acc = cb[d];
  #pragma unroll
  for (int j = 0; j < DCONV; ++j) {
    const int ts = t - (DCONV - 1) + j;
    if (ts >= 0)
      acc += cw[d * DCONV + j] * xz[((size_t)b * SEQ + ts) * (2 * DINNER) + d];
  }
  u[idx] = acc / (1.f + __expf(-acc));   // SiLU
}

// ---------------------------------------------------------------------------
// Selective scan, one lane per (channel, state): 16 lanes = 1 channel,
// sum over states via 4x shfl_xor (stays inside 16-lane groups on wave32).
// Fused epilogue (lane n==0): y = (scan + u*D) * silu(z).
// B_t / C_t (shared by all channels) staged in LDS in 128-step chunks.
// grid = (DINNER/16, BATCH), block = 256 (= 16 channels).
// ---------------------------------------------------------------------------
__global__ void scan_gate(const float* __restrict__ dt, const float* __restrict__ u,
                          const float* __restrict__ xdbl, const float* __restrict__ xz,
                          const float* __restrict__ A_log, const float* __restrict__ Dvec,
                          float* __restrict__ yg)
{
  const int tid = threadIdx.x;
  const int n   = tid & 15;        // state index
  const int ch  = tid >> 4;        // channel within block (0..15)
  const int d   = blockIdx.x * 16 + ch;
  const int b   = blockIdx.y;

  __shared__ float sB[SCHUNK * DSTATE];
  __shared__ float sC[SCHUNK * DSTATE];

  const float Ad = -__expf(A_log[d * DSTATE + n]);
  float h = 0.f;

  for (int t0 = 0; t0 < SEQ; t0 += SCHUNK) {
    for (int i = tid; i < SCHUNK * DSTATE; i += 256) {
      const size_t base = ((size_t)b * SEQ + t0 + (i >> 4)) * (DTRANK + 2 * DSTATE);
      sB[i] = xdbl[base + DTRANK + (i & 15)];
      sC[i] = xdbl[base + DTRANK + DSTATE + (i & 15)];
    }
    __syncthreads();

    for (int tt = 0; tt < SCHUNK; ++tt) {
      const int t = t0 + tt;
      const size_t idx = ((size_t)b * SEQ + t) * DINNER + d;
      const float dtv = dt[idx];                 // broadcast within 16-lane group
      const float uv  = u[idx];
      h = __expf(dtv * Ad) * h + dtv * sB[tt * DSTATE + n] * uv;
      float p = h * sC[tt * DSTATE + n];
      p += __shfl_xor(p, 1);
      p += __shfl_xor(p, 2);
      p += __shfl_xor(p, 4);
      p += __shfl_xor(p, 8);
      if (n == 0) {
        const float zv = xz[((size_t)b * SEQ + t) * (2 * DINNER) + DINNER + d];
        yg[idx] = (p + uv * Dvec[d]) * (zv / (1.f + __expf(-zv)));
      }
    }
    __syncthreads();
  }
}

// ---------------------------------------------------------------------------
// res = out + x, then LayerNorm over DMODEL. One block (256 thr) per row.
// ---------------------------------------------------------------------------
__global__ void layernorm_residual(const float* __restrict__ om, const float* __restrict__ x,
                                   const float* __restrict__ gamma, const float* __restrict__ beta,
                                   float* __restrict__ y)
{
  const int row = blockIdx.x;
  const int tid = threadIdx.x;
  __shared__ float srow[DMODEL];
  __shared__ float sred[256];

  float s = 0.f;
  for (int i = tid; i < DMODEL; i += 256) {
    const float v = om[(size_t)row * DMODEL + i] + x[(size_t)row * DMODEL + i];
    srow[i] = v; s += v;
  }
  sred[tid] = s; __syncthreads();
  for (int off = 128; off > 0; off >>= 1) {
    if (tid < off) sred[tid] += sred[tid + off];
    __syncthreads();
  }
  const float mu = sred[0] * (1.f / DMODEL);
  __syncthreads();

  float s2 = 0.f;
  for (int i = tid; i < DMODEL; i += 256) {
    const float dv = srow[i] - mu; s2 += dv * dv;
  }
  sred[tid] = s2; __syncthreads();
  for (int off = 128; off > 0; off >>= 1) {
    if (tid < off) sred[tid] += sred[tid + off];
    __syncthreads();
  }
  const float inv = rsqrtf(sred[0] * (1.f / DMODEL) + LNEPS);

  for (int i = tid; i < DMODEL; i += 256)
    y[(size_t)row * DMODEL + i] = (srow[i] - mu) * inv * gamma[i] + beta[i];
}

// ---------------------------------------------------------------------------
extern "C" void kernel_launch(void* const* d_in, const int* in_sizes, int n_in,
                              void* d_out, int out_size, void* d_ws, size_t ws_size,
                              hipStream_t stream) {
  (void)in_sizes; (void)n_in; (void)out_size; (void)ws_size;
  const float* x      = (const float*)d_in[0];
  const float* W_in   = (const float*)d_in[1];
  const float* conv_w = (const float*)d_in[2];
  const float* conv_b = (const float*)d_in[3];
  const float* W_xprj = (const float*)d_in[4];
  const float* W_dt   = (const float*)d_in[5];
  const float* b_dt   = (const float*)d_in[6];
  const float* A_log  = (const float*)d_in[7];
  const float* Dvec   = (const float*)d_in[8];
  const float* W_out  = (const float*)d_in[9];
  const float* gamma  = (const float*)d_in[10];
  const float* beta   = (const float*)d_in[11];
  float* out = (float*)d_out;

  const int M = BATCH * SEQ;                     // 4096 token rows
  float* ws   = (float*)d_ws;
  float* xz   = ws;  ws += (size_t)M * (2 * DINNER);          // [M, 4096]
  float* u    = ws;  ws += (size_t)M * DINNER;                // [M, 2048]
  float* xdbl = ws;  ws += (size_t)M * (DTRANK + 2 * DSTATE); // [M, 96]
  float* dt   = ws;  ws += (size_t)M * DINNER;                // [M, 2048]
  float* yg   = ws;  ws += (size_t)M * DINNER;                // [M, 2048]
  float* om   = ws;  ws += (size_t)M * DMODEL;                // [M, 1024]

  const dim3 blk(256);

  // 1) xz = x @ W_in                       [M,1024] x [1024,4096]
  wmma_gemm_f32<DMODEL, DMODEL, 2*DINNER, 2*DINNER, 0>
      <<<dim3((2 * DINNER + 127) / 128, M / 16), blk, 0, stream>>>(
      x, W_in, xz, nullptr, 2 * DINNER);

  // 2) depthwise causal conv + SiLU -> u
  conv_silu<<<dim3(((size_t)M * DINNER + 255) / 256), blk, 0, stream>>>(
      xz, conv_w, conv_b, u);

  // 3) x_dbl = u @ W_xproj                 [M,2048] x [2048,96]
  wmma_gemm_f32<DINNER, DINNER, DTRANK + 2*DSTATE, DTRANK + 2*DSTATE, 0>
      <<<dim3(1, M / 16), blk, 0, stream>>>(
      u, W_xprj, xdbl, nullptr, DTRANK + 2 * DSTATE);

  // 4) dt = softplus(x_dbl[:, :64] @ W_dt + b_dt)   [M,64] x [64,2048]
  wmma_gemm_f32<DTRANK, DTRANK + 2*DSTATE, DINNER, DINNER, 1>
      <<<dim3((DINNER + 127) / 128, M / 16), blk, 0, stream>>>(
      xdbl, W_dt, dt, b_dt, DINNER);

  // 5) selective scan + gating -> yg
  scan_gate<<<dim3(DINNER / 16, BATCH), blk, 0, stream>>>(
      dt, u, xdbl, xz, A_log, Dvec, yg);

  // 6) om = yg @ W_out                     [M,2048] x [2048,1024]
  wmma_gemm_f32<DINNER, DINNER, DMODEL, DMODEL, 0>
      <<<dim3((DMODEL + 127) / 128, M / 16), blk, 0, stream>>>(
      yg, W_out, om, nullptr, DMODEL);

  // 7) residual + LayerNorm -> d_out
  layernorm_residual<<<dim3(M), blk, 0, stream>>>(om, x, gamma, beta, out);
}